// YOLO1_58720792871045
// MI455X (gfx1250) — compile-verified
//
#include <hip/hip_runtime.h>
#include <stdint.h>

#define S_CELLS   196
#define SPAD      256
#define CFEAT     30
#define NCLS      20
#define THR_IOU   0.8f
#define BG_CLASS  20
#define CLS_OFF   10000.0f

__device__ __forceinline__ unsigned f2sortable(float f) {
    unsigned u = __float_as_uint(f);
    return (u & 0x80000000u) ? ~u : (u | 0x80000000u);
}

__device__ __forceinline__ float sigmoidf_(float x) {
    return 1.0f / (1.0f + __expf(-x));
}

__global__ __launch_bounds__(256) void yolo_decode_nms(
    const float* __restrict__ features,
    const int* __restrict__ p_hori, const int* __restrict__ p_wori,
    const int* __restrict__ p_hfea, const int* __restrict__ p_wfea,
    float* __restrict__ out, int B)
{
    // 196*30 = 5880 floats used; padded to 1536 * 16B chunks so every lane of
    // every async-load instruction is active (tail lanes dump into the pad).
    __shared__ float              featL[6144];
    __shared__ float              oboxL[S_CELLS * 4];
    __shared__ float              probL[S_CELLS];
    __shared__ int                catL[S_CELLS];
    __shared__ int                validL[S_CELLS];
    __shared__ unsigned long long keyL[SPAD];
    __shared__ float              sboxL[SPAD * 4];
    __shared__ float              sareaL[SPAD];
    __shared__ int                skeepL[SPAD];
    __shared__ int                sidxL[SPAD];
    __shared__ int                keepOrig[S_CELLS];

    const int tid   = (int)threadIdx.x;
    const int batch = (int)blockIdx.x;

    // ---------------- Stage 0: async global -> LDS copy of this image ----------------
    const unsigned bytesValid = S_CELLS * CFEAT * 4u;   // 23520
    unsigned long long gbase = (unsigned long long)(uintptr_t)features
                             + (unsigned long long)batch * (unsigned long long)bytesValid;
    // Low 32 bits of a generic pointer to LDS == byte offset within the WG's LDS alloc.
    unsigned ldsBase = (unsigned)(unsigned long long)(uintptr_t)(&featL[0]);

#pragma unroll
    for (int r = 0; r < 6; ++r) {
        unsigned c    = (unsigned)tid + (unsigned)r * 256u;   // 16B chunk id, 0..1535
        unsigned goff = c * 16u;
        if (goff >= bytesValid) goff = 0u;                    // clamp tail (re-read head)
        unsigned loff = ldsBase + c * 16u;
        // GVS mode: mem = SADDR(sgpr64) + VADDR(vgpr32); LDS dest = VDST(vgpr32)
        asm volatile("global_load_async_to_lds_b128 %0, %1, %2"
                     :
                     : "v"(loff), "v"(goff), "s"(gbase)
                     : "memory");
    }
    asm volatile("s_wait_asynccnt 0x0" ::: "memory");
    __syncthreads();

    const int h_ori = *p_hori, w_ori = *p_wori;
    const int h_fea = *p_hfea, w_fea = *p_wfea;

    // ---------------- Stage 1: per-cell decode ----------------
    if (tid < S_CELLS) {
        const float* f = &featL[tid * CFEAT];

        float c4 = f[4], c9 = f[9];
        float cmax = fmaxf(c4, c9);
        int   carg = (c9 > c4) ? 1 : 0;       // first-max semantics of argmax

        // softmax over classes (max-subtracted, like jax.nn.softmax)
        float mx = f[10];
#pragma unroll
        for (int k = 1; k < NCLS; ++k) mx = fmaxf(mx, f[10 + k]);
        float e[NCLS];
        float sum = 0.0f;
#pragma unroll
        for (int k = 0; k < NCLS; ++k) { e[k] = __expf(f[10 + k] - mx); sum += e[k]; }
        float inv = 1.0f / sum;

        float best = cmax * (e[0] * inv);
        int   cat  = 0;
#pragma unroll
        for (int k = 1; k < NCLS; ++k) {
            float v = cmax * (e[k] * inv);
            if (v > best) { best = v; cat = k; }   // first-max on ties
        }

        int b0 = carg * 5;
        float bx = sigmoidf_(f[b0 + 0]);
        float by = sigmoidf_(f[b0 + 1]);
        float bw = sigmoidf_(f[b0 + 2]);
        float bh = sigmoidf_(f[b0 + 3]);

        int rowc = tid / w_fea;
        int colc = tid - rowc * w_fea;
        float sx = (float)w_ori / (float)w_fea;
        float sy = (float)h_ori / (float)h_fea;
        float cx = ((float)colc + bx) * sx;
        float cy = ((float)rowc + by) * sy;
        float BW = bw * (float)w_ori;
        float BH = bh * (float)h_ori;

        float x1 = cx - BW * 0.5f, y1 = cy - BH * 0.5f;
        float x2 = cx + BW * 0.5f, y2 = cy + BH * 0.5f;

        oboxL[tid * 4 + 0] = x1;
        oboxL[tid * 4 + 1] = y1;
        oboxL[tid * 4 + 2] = x2;
        oboxL[tid * 4 + 3] = y2;
        probL[tid]  = best;
        catL[tid]   = cat;
        int valid   = (cat != BG_CLASS);
        validL[tid] = valid;

        float sc = valid ? best : -INFINITY;
        keyL[tid] = (((unsigned long long)f2sortable(sc)) << 32) | (unsigned)(~tid);
    } else {
        keyL[tid] = (((unsigned long long)f2sortable(-INFINITY)) << 32) | (unsigned)(~tid);
    }
    __syncthreads();

    // ---------------- Stage 2: bitonic sort of 256 keys, descending ----------------
    // (emulates stable argsort(-score): key = score, tiebreak ~idx => idx ascending)
    for (unsigned k = 2; k <= SPAD; k <<= 1) {
        for (unsigned j = k >> 1; j > 0; j >>= 1) {
            __syncthreads();
            unsigned i   = (unsigned)tid;
            unsigned ixj = i ^ j;
            if (ixj > i) {
                unsigned long long a = keyL[i];
                unsigned long long b = keyL[ixj];
                bool descSeg = ((i & k) == 0);
                bool doSwap  = descSeg ? (a < b) : (a > b);
                if (doSwap) { keyL[i] = b; keyL[ixj] = a; }
            }
        }
    }
    __syncthreads();

    // ---------------- Stage 3: gather into sorted order (with class offset) -------
    {
        int o = (int)(~((unsigned)keyL[tid]));   // original cell index (0..255)
        sidxL[tid] = o;
        if (o < S_CELLS) {
            skeepL[tid] = validL[o];
            float co = (float)catL[o] * CLS_OFF;
            float x1 = oboxL[o * 4 + 0] + co;
            float y1 = oboxL[o * 4 + 1] + co;
            float x2 = oboxL[o * 4 + 2] + co;
            float y2 = oboxL[o * 4 + 3] + co;
            sboxL[tid * 4 + 0] = x1;
            sboxL[tid * 4 + 1] = y1;
            sboxL[tid * 4 + 2] = x2;
            sboxL[tid * 4 + 3] = y2;
            sareaL[tid] = fmaxf(x2 - x1, 0.0f) * fmaxf(y2 - y1, 0.0f);
        } else {
            skeepL[tid] = 0;
            sboxL[tid * 4 + 0] = 0.0f;
            sboxL[tid * 4 + 1] = 0.0f;
            sboxL[tid * 4 + 2] = 0.0f;
            sboxL[tid * 4 + 3] = 0.0f;
            sareaL[tid] = 0.0f;
        }
    }

    // ---------------- Stage 4: greedy NMS sweep (sequential in i, parallel in j) ---
    for (int i = 0; i < S_CELLS; ++i) {
        __syncthreads();
        if (skeepL[i] == 0) continue;            // uniform branch
        int j = tid;
        if (j > i && skeepL[j]) {
            float xi1 = sboxL[i * 4 + 0], yi1 = sboxL[i * 4 + 1];
            float xi2 = sboxL[i * 4 + 2], yi2 = sboxL[i * 4 + 3];
            float xj1 = sboxL[j * 4 + 0], yj1 = sboxL[j * 4 + 1];
            float xj2 = sboxL[j * 4 + 2], yj2 = sboxL[j * 4 + 3];
            float ix1 = fmaxf(xi1, xj1), iy1 = fmaxf(yi1, yj1);
            float ix2 = fminf(xi2, xj2), iy2 = fminf(yi2, yj2);
            float inter = fmaxf(ix2 - ix1, 0.0f) * fmaxf(iy2 - iy1, 0.0f);
            float uni   = sareaL[i] + sareaL[j] - inter;
            float iou   = inter / fmaxf(uni, 1e-9f);
            if (iou > THR_IOU) skeepL[j] = 0;
        }
    }
    __syncthreads();

    // ---------------- Stage 5: scatter keep back to original cell order -----------
    {
        int o = sidxL[tid];
        if (o < S_CELLS) keepOrig[o] = skeepL[tid];
    }
    __syncthreads();

    // ---------------- Stage 6: write outputs (boxes*m, prob*m, cat, keep) ---------
    if (tid < S_CELLS) {
        long long cell = (long long)batch * S_CELLS + tid;
        long long BS   = (long long)B * S_CELLS;
        float m = keepOrig[tid] ? 1.0f : 0.0f;

        float* ob = out + cell * 4;
        ob[0] = oboxL[tid * 4 + 0] * m;
        ob[1] = oboxL[tid * 4 + 1] * m;
        ob[2] = oboxL[tid * 4 + 2] * m;
        ob[3] = oboxL[tid * 4 + 3] * m;
        out[BS * 4 + cell] = probL[tid] * m;
        out[BS * 5 + cell] = (float)catL[tid];
        out[BS * 6 + cell] = m;
    }
}

extern "C" void kernel_launch(void* const* d_in, const int* in_sizes, int n_in,
                              void* d_out, int out_size, void* d_ws, size_t ws_size,
                              hipStream_t stream) {
    (void)in_sizes; (void)n_in; (void)d_ws; (void)ws_size;
    const float* features = (const float*)d_in[0];
    const int*   h_ori    = (const int*)d_in[1];
    const int*   w_ori    = (const int*)d_in[2];
    const int*   h_fea    = (const int*)d_in[3];
    const int*   w_fea    = (const int*)d_in[4];
    float*       out      = (float*)d_out;

    // out = boxes(B*S*4) + prob(B*S) + cat(B*S) + keep(B*S) = 7*B*S floats
    int cells = out_size / 7;
    int B     = cells / S_CELLS;
    if (B <= 0) return;

    yolo_decode_nms<<<dim3((unsigned)B), dim3(256), 0, stream>>>(
        features, h_ori, w_ori, h_fea, w_fea, out, B);
}